// f_NSSM_54142357733381
// MI455X (gfx1250) — compile-verified
//
#include <hip/hip_runtime.h>

typedef __attribute__((ext_vector_type(2))) float v2f;
typedef __attribute__((ext_vector_type(8))) float v8f;

#define T_STEPS 2048
#define BATCH   64
#define S_DIM   256
#define IN_DIM  64
#define DIST_DIM 32

__device__ __forceinline__ v8f wmma_f32(v2f a, v2f b, v8f c) {
  // V_WMMA_F32_16X16X4_F32: D = A(16x4) * B(4x16) + C(16x16), fp32 throughout.
  return __builtin_amdgcn_wmma_f32_16x16x4_f32(
      /*neg_a=*/false, a, /*neg_b=*/false, b,
      /*c_mod=*/(short)0, c, /*reuse_a=*/false, /*reuse_b=*/false);
}

// ---------------------------------------------------------------------------
// Phase 1: drive[t,b,:] = U[t,b,:] @ B_w^T + D[t,b,:] @ E_w^T
// One wave per 16x16 output tile of the [T*B, S] drive matrix.
// Result is staged into the first X copy of d_out (scan consumes + overwrites).
// ---------------------------------------------------------------------------
__global__ __launch_bounds__(256) void nssm_drive_kernel(
    const float* __restrict__ U, const float* __restrict__ D,
    const float* __restrict__ Bw, const float* __restrict__ Ew,
    float* __restrict__ outX) {
  const int tid  = threadIdx.x;
  const int wave = tid >> 5;
  const int lane = tid & 31;
  const int half = lane >> 4;   // which 16-lane half of the wave
  const int mn   = lane & 15;   // M index for A/C rows, N index for B/C cols

  const int tile     = blockIdx.x * 8 + wave;   // 131072 tiles total
  const int row_tile = tile >> 4;               // 8192 row tiles over T*B
  const int col_tile = tile & 15;               // 16 col tiles over S
  const int row0 = row_tile * 16;
  const int col0 = col_tile * 16;
  const int n    = col0 + mn;

  v8f c = {};
#pragma unroll
  for (int kk = 0; kk < IN_DIM / 4; ++kk) {          // K = 64 over U / B_w
    const int k2 = 4 * kk + 2 * half;
    v2f a = *(const v2f*)(U  + (size_t)(row0 + mn) * IN_DIM + k2);
    v2f b = *(const v2f*)(Bw + (size_t)n * IN_DIM + k2);
    c = wmma_f32(a, b, c);
  }
#pragma unroll
  for (int kk = 0; kk < DIST_DIM / 4; ++kk) {        // K = 32 over D / E_w
    const int k2 = 4 * kk + 2 * half;
    v2f a = *(const v2f*)(D  + (size_t)(row0 + mn) * DIST_DIM + k2);
    v2f b = *(const v2f*)(Ew + (size_t)n * DIST_DIM + k2);
    c = wmma_f32(a, b, c);
  }

  const size_t base = (size_t)row0 * S_DIM + col0;
#pragma unroll
  for (int i = 0; i < 8; ++i)
    outX[base + (size_t)(i + 8 * half) * S_DIM + mn] = c[i];
}

// ---------------------------------------------------------------------------
// Phase 2: sequential scan  x[t] = x[t-1] @ A_w^T + drive[t]
// 4 workgroups, each owning 16 independent batch rows (M tile = 16).
// 16 waves/WG: wave w owns output columns [16w, 16w+16) and holds the
// corresponding A_w^T[256 x 16] slice in 64 resident B-fragments (128 VGPRs).
// State ping-pongs in LDS (32 KB); one barrier per step.
// ---------------------------------------------------------------------------
__global__ __launch_bounds__(512) void nssm_scan_kernel(
    const float* __restrict__ x0, const float* __restrict__ Aw,
    float* __restrict__ outX, float* __restrict__ outY) {
  __shared__ float xbuf[2][16 * S_DIM];   // 2 x 16 rows x 256 cols = 32 KB

  const int tid  = threadIdx.x;
  const int wave = tid >> 5;
  const int lane = tid & 31;
  const int half = lane >> 4;
  const int mn   = lane & 15;
  const int g    = blockIdx.x;            // batch group: rows [16g, 16g+16)
  const int col0 = wave * 16;
  const int n    = col0 + mn;

  // Load this wave's A_w^T slice once; lives in VGPRs for all 2048 steps.
  v2f bfrag[S_DIM / 4];
#pragma unroll
  for (int kk = 0; kk < S_DIM / 4; ++kk)
    bfrag[kk] = *(const v2f*)(Aw + (size_t)n * S_DIM + 4 * kk + 2 * half);

  // Stage x0 rows for this batch group into LDS buffer 0.
  for (int i = tid; i < 16 * S_DIM; i += 512)
    xbuf[0][i] = x0[(size_t)(g * 16 + (i >> 8)) * S_DIM + (i & 255)];
  __syncthreads();

  for (int t = 0; t < T_STEPS; ++t) {
    const int cur = t & 1, nxt = cur ^ 1;
    const size_t base =
        (size_t)t * (BATCH * S_DIM) + (size_t)g * 16 * S_DIM + col0;

    // C init = drive[t] tile (staged in outX by phase 1).
    v8f c;
#pragma unroll
    for (int i = 0; i < 8; ++i)
      c[i] = outX[base + (size_t)(i + 8 * half) * S_DIM + mn];

    // Prefetch next step's drive tile while WMMAs run.
    if (t + 1 < T_STEPS)
      __builtin_prefetch(outX + base + (size_t)BATCH * S_DIM +
                             (size_t)(8 * half) * S_DIM + mn, 0, 1);

    // x_new tile = x_prev(16x256) @ A_w^T slice(256x16): 64 K-steps of 4.
#pragma unroll
    for (int kk = 0; kk < S_DIM / 4; ++kk) {
      v2f a = *(const v2f*)&xbuf[cur][mn * S_DIM + 4 * kk + 2 * half];
      c = wmma_f32(a, bfrag[kk], c);
    }

    // Publish x[t]: LDS (for next step) + both output copies.
#pragma unroll
    for (int i = 0; i < 8; ++i) {
      const int m = i + 8 * half;
      xbuf[nxt][m * S_DIM + col0 + mn] = c[i];
      outX[base + (size_t)m * S_DIM + mn] = c[i];
      outY[base + (size_t)m * S_DIM + mn] = c[i];
    }
    __syncthreads();
  }
}

// ---------------------------------------------------------------------------
extern "C" void kernel_launch(void* const* d_in, const int* in_sizes, int n_in,
                              void* d_out, int out_size, void* d_ws,
                              size_t ws_size, hipStream_t stream) {
  (void)in_sizes; (void)n_in; (void)out_size; (void)d_ws; (void)ws_size;
  const float* x0 = (const float*)d_in[0];
  const float* U  = (const float*)d_in[1];
  const float* D  = (const float*)d_in[2];
  const float* Aw = (const float*)d_in[3];
  const float* Bw = (const float*)d_in[4];
  const float* Ew = (const float*)d_in[5];

  float* outX = (float*)d_out;                                   // X copy 1
  float* outY = outX + (size_t)T_STEPS * BATCH * S_DIM;          // X copy 2

  const int total_tiles = (T_STEPS * BATCH / 16) * (S_DIM / 16); // 131072
  nssm_drive_kernel<<<total_tiles / 8, 256, 0, stream>>>(U, D, Bw, Ew, outX);
  nssm_scan_kernel<<<4, 512, 0, stream>>>(x0, Aw, outX, outY);
}